// CausalDepthwiseConv1d_49074296324237
// MI455X (gfx1250) — compile-verified
//
#include <hip/hip_runtime.h>

// Problem constants (from reference): x[B][L][D], weight[D][K], bias[D], K=4 causal depthwise.
#define B_ 4
#define L_ 4096
#define D_ 4096
#define K_ 4

// Tiling
#define DT 512            // channels per block tile (2 per thread)
#define TL 4              // L-rows staged per async step
#define RING 16           // LDS ring rows (pow2). Live span = 7 (compute) + 8 (2 in-flight tiles) = 15 <= 16
#define LC 128            // L-rows per block chunk
#define NT (LC / TL)      // async steps per block
#define NTHREADS 256      // 8 wave32

// True LDS (addrspace 3) pointer type: 32-bit, value == LDS byte offset seen by hardware.
typedef __attribute__((address_space(3))) float lds_f;

// ---- CDNA5 async global->LDS copy engine (ASYNCcnt-tracked), via inline asm ----
// The LDS destination is passed as a genuine AS3 pointer derived from smem:
//  (a) its integer value is the exact hardware LDS offset (no base-0 assumption),
//  (b) smem escapes into the asm, so the compiler must keep the ds_load reads
//      of data the asm deposits (previously GlobalOpt folded them to undef).
__device__ __forceinline__ void async_ld_b128(lds_f* laddr, const float* gaddr) {
    asm volatile("global_load_async_to_lds_b128 %0, %1, off"
                 :: "v"(laddr), "v"(gaddr) : "memory");
}
__device__ __forceinline__ void async_ld_b64(lds_f* laddr, const float* gaddr) {
    asm volatile("global_load_async_to_lds_b64 %0, %1, off"
                 :: "v"(laddr), "v"(gaddr) : "memory");
}
// Split dependency counters (gfx1250): wait on ASYNCcnt only.
__device__ __forceinline__ void wait_async_le2() { asm volatile("s_wait_asynccnt 0x2" ::: "memory"); }
__device__ __forceinline__ void wait_async_0()   { asm volatile("s_wait_asynccnt 0x0" ::: "memory"); }

__global__ __launch_bounds__(NTHREADS)
void causal_dwconv1d_kernel(const float* __restrict__ x,
                            const float* __restrict__ weight,
                            const float* __restrict__ bias,
                            float* __restrict__ out) {
    __shared__ float smem[RING * DT];   // 32 KB ring buffer
    lds_f* sm3 = (lds_f*)smem;          // addrspacecast generic -> AS3 (escapes smem)

    const int tid = threadIdx.x;
    const int nDT = D_ / DT;            // 8
    const int nLC = L_ / LC;            // 32

    int bid = blockIdx.x;
    const int b    = bid / (nDT * nLC);
    int rem        = bid % (nDT * nLC);
    const int dti  = rem / nLC;
    const int lci  = rem % nLC;
    const int d0   = dti * DT;
    const int lstart = lci * LC;

    const size_t xbase = ((size_t)b * L_) * (size_t)D_ + (size_t)d0;  // &x[b][0][d0]

    // Per-thread channel pair and its weights/bias (weight is [D][4] -> one float4 per channel).
    const int c0 = d0 + 2 * tid;
    const float4 wA = ((const float4*)weight)[c0];
    const float4 wB = ((const float4*)weight)[c0 + 1];
    const float2 bb = *(const float2*)(bias + c0);
    const float* wAp = &wA.x;
    const float* wBp = &wB.x;

    // ---- Prologue: stage halo rows (lstart-3..lstart-1) when chunk is interior ----
    if (lci > 0) {
        #pragma unroll
        for (int r = 0; r < K_ - 1; ++r) {
            const int lt = lstart - (K_ - 1) + r;
            async_ld_b64(sm3 + ((lt & (RING - 1)) * DT + 2 * tid),
                         x + xbase + (size_t)lt * D_ + 2 * tid);
        }
    }
    // ---- Prologue: stage tile 0 (rows lstart..lstart+3), 2 x b128 per thread, coalesced ----
    {
        #pragma unroll
        for (int q = 0; q < 2; ++q) {
            const int j   = tid + q * NTHREADS;     // 0..511 float4 slots of the tile
            const int row = j >> 7;                 // DT/4 = 128 float4 per row
            const int col = (j & 127) << 2;
            const int lt  = lstart + row;
            async_ld_b128(sm3 + ((lt & (RING - 1)) * DT + col),
                          x + xbase + (size_t)lt * D_ + col);
        }
    }

    // ---- Pipelined main loop: issue tile t+1, wait tile t, barrier, compute tile t ----
    for (int t = 0; t < NT; ++t) {
        const int l0 = lstart + t * TL;

        if (t + 1 < NT) {
            const int l0n = l0 + TL;
            #pragma unroll
            for (int q = 0; q < 2; ++q) {
                const int j   = tid + q * NTHREADS;
                const int row = j >> 7;
                const int col = (j & 127) << 2;
                const int lt  = l0n + row;
                async_ld_b128(sm3 + ((lt & (RING - 1)) * DT + col),
                              x + xbase + (size_t)lt * D_ + col);
            }
            wait_async_le2();   // tile t resident; tile t+1 (2 async ops) may remain in flight
        } else {
            wait_async_0();
        }
        __syncthreads();        // all waves' slices of tile t visible in LDS

        // 7-row sliding window (rows l0-3 .. l0+3); causal zero-pad for l < 0.
        float2 xs[TL + K_ - 1];
        #pragma unroll
        for (int j = 0; j < TL + K_ - 1; ++j) {
            const int lt = l0 - (K_ - 1) + j;
            if (lt >= 0) {
                xs[j] = *(const float2*)&smem[(lt & (RING - 1)) * DT + 2 * tid];
            } else {
                xs[j] = make_float2(0.0f, 0.0f);
            }
        }

        #pragma unroll
        for (int i = 0; i < TL; ++i) {
            float2 acc = bb;
            #pragma unroll
            for (int k = 0; k < K_; ++k) {
                acc.x = fmaf(xs[i + k].x, wAp[k], acc.x);
                acc.y = fmaf(xs[i + k].y, wBp[k], acc.y);
            }
            const int l = l0 + i;
            *(float2*)(out + xbase + (size_t)l * D_ + 2 * tid) = acc;
        }
        // No second barrier needed: with RING=16, slots written by tiles t+1/t+2
        // are disjoint from the rows any wave may still be reading for tile t.
    }
}

extern "C" void kernel_launch(void* const* d_in, const int* in_sizes, int n_in,
                              void* d_out, int out_size, void* d_ws, size_t ws_size,
                              hipStream_t stream) {
    const float* x      = (const float*)d_in[0];
    const float* weight = (const float*)d_in[1];
    const float* bias   = (const float*)d_in[2];
    float* out          = (float*)d_out;

    const int grid = B_ * (D_ / DT) * (L_ / LC);   // 4 * 8 * 32 = 1024 blocks
    causal_dwconv1d_kernel<<<grid, NTHREADS, 0, stream>>>(x, weight, bias, out);
}